// VenomGNN_75479755260529
// MI455X (gfx1250) — compile-verified
//
#include <hip/hip_runtime.h>
#include <math.h>

#define NEG_SLOPE 0.2f
#define EPS_GAT 1e-16f

typedef __attribute__((ext_vector_type(2))) float v2f;
typedef __attribute__((ext_vector_type(8))) float v8f;

// ---------------------------------------------------------------------------
// FP32 WMMA GEMM: C[M,Ncols] = A[M,K] @ B[K,Ncols]   (row-major, M%16==0 ok,
// K%4==0, Ncols%16==0). One wave32 computes one 16x16 tile using
// V_WMMA_F32_16X16X4_F32 across K in steps of 4.
// Fragment layouts per CDNA5 ISA 7.12.2:
//   A 16x4: lanes 0-15 -> M=lane, regs {K0,K1}; lanes 16-31 -> regs {K2,K3}
//   B 4x16: lanes 0-15 -> N=lane, regs {K0 row, K1 row}; lanes 16-31 -> {K2,K3}
//   C 16x16: VGPR v, lanes 0-15 -> M=v, N=lane; lanes 16-31 -> M=v+8
// ---------------------------------------------------------------------------
__global__ void wmma_gemm_f32_k4(const float* __restrict__ A,
                                 const float* __restrict__ B,
                                 float* __restrict__ C,
                                 int M, int K, int Ncols) {
  const int wave = (int)((blockIdx.x * blockDim.x + threadIdx.x) >> 5);
  const int lane = (int)(threadIdx.x & 31);
  const int ntiles = Ncols >> 4;
  const int mt = wave / ntiles;
  const int nt = wave - mt * ntiles;
  if (mt * 16 >= M) return;              // wave-uniform: EXEC stays all-ones
  const int half = lane >> 4;            // 0 or 1
  const int l15 = lane & 15;
  const long arow = (long)(mt * 16 + l15) * (long)K;
  const int col = nt * 16 + l15;
  v8f acc = {};
  for (int kb = 0; kb < K; kb += 4) {
    const int k0 = kb + half * 2;
    v2f a, b;
    a.x = A[arow + k0];
    a.y = A[arow + k0 + 1];
    b.x = B[(long)k0 * Ncols + col];
    b.y = B[(long)(k0 + 1) * Ncols + col];
    acc = __builtin_amdgcn_wmma_f32_16x16x4_f32(false, a, false, b,
                                                (short)0, acc, false, false);
  }
  const int crow0 = mt * 16 + half * 8;
#pragma unroll
  for (int v = 0; v < 8; ++v)
    C[(long)(crow0 + v) * Ncols + col] = acc[v];
}

// ---------------------------------------------------------------------------
// alpha_s[n,h] = dot(h[n, h*64 : h*64+64], a_src[h]);  same for alpha_d.
// ---------------------------------------------------------------------------
__global__ void alpha_kernel(const float* __restrict__ h,
                             const float* __restrict__ a_src,
                             const float* __restrict__ a_dst,
                             float* __restrict__ as, float* __restrict__ ad,
                             int N, int H) {
  int idx = blockIdx.x * blockDim.x + threadIdx.x;
  if (idx >= N * H) return;
  int n = idx / H, hd = idx - n * H;
  const float* row = h + (long)n * H * 64 + hd * 64;
  const float* vs = a_src + hd * 64;
  const float* vd = a_dst + hd * 64;
  float ss = 0.f, sd = 0.f;
#pragma unroll 4
  for (int c = 0; c < 64; ++c) {
    float x = row[c];
    ss += x * vs[c];
    sd += x * vd[c];
  }
  as[idx] = ss;
  ad[idx] = sd;
}

__device__ __forceinline__ void atomicMaxFloat(float* addr, float val) {
  if (val >= 0.f)
    atomicMax((int*)addr, __float_as_int(val));
  else
    atomicMin((unsigned int*)addr, __float_as_uint(val));
}

__device__ __forceinline__ void edge_sd(const int* __restrict__ ei, int E,
                                        int e, int& s, int& d) {
  if (e < E) { s = ei[e]; d = ei[E + e]; }
  else       { s = e - E; d = e - E; }     // self loop
}

// pass 1: segment max of leaky_relu(alpha_s[src]+alpha_d[dst]) over dst
__global__ void edge_max_kernel(const int* __restrict__ ei, int E, int N,
                                const float* __restrict__ as,
                                const float* __restrict__ ad,
                                float* __restrict__ maxb, int H) {
  int e = blockIdx.x * blockDim.x + threadIdx.x;
  if (e >= E + N) return;
  int s, d;
  edge_sd(ei, E, e, s, d);
  for (int h = 0; h < H; ++h) {
    float el = as[s * H + h] + ad[d * H + h];
    el = el > 0.f ? el : NEG_SLOPE * el;
    atomicMaxFloat(&maxb[d * H + h], el);
  }
}

// pass 2: segment sum of exp(e - max[dst])
__global__ void edge_expsum_kernel(const int* __restrict__ ei, int E, int N,
                                   const float* __restrict__ as,
                                   const float* __restrict__ ad,
                                   const float* __restrict__ maxb,
                                   float* __restrict__ sumb, int H) {
  int e = blockIdx.x * blockDim.x + threadIdx.x;
  if (e >= E + N) return;
  int s, d;
  edge_sd(ei, E, e, s, d);
  for (int h = 0; h < H; ++h) {
    float el = as[s * H + h] + ad[d * H + h];
    el = el > 0.f ? el : NEG_SLOPE * el;
    atomicAdd(&sumb[d * H + h], expf(el - maxb[d * H + h]));
  }
}

// pass 3: out[dst] += alpha * h[src]; 64 threads per edge (one per channel),
// each thread walks all H heads. alpha recompute is VALU-cheap; the atomicAdd
// RMW lands in L2 (out fits in the 192MB L2).
__global__ void edge_aggregate_kernel(const int* __restrict__ ei, int E, int N,
                                      const float* __restrict__ as,
                                      const float* __restrict__ ad,
                                      const float* __restrict__ maxb,
                                      const float* __restrict__ sumb,
                                      const float* __restrict__ h,
                                      float* __restrict__ out, int H) {
  int t = blockIdx.x * blockDim.x + threadIdx.x;
  int e = t >> 6;
  int c = t & 63;
  if (e >= E + N) return;
  int s, d;
  edge_sd(ei, E, e, s, d);
  for (int hd = 0; hd < H; ++hd) {
    float el = as[s * H + hd] + ad[d * H + hd];
    el = el > 0.f ? el : NEG_SLOPE * el;
    float alpha = expf(el - maxb[d * H + hd]) / (sumb[d * H + hd] + EPS_GAT);
    atomicAdd(&out[(long)d * H * 64 + hd * 64 + c],
              alpha * h[(long)s * H * 64 + hd * 64 + c]);
  }
}

__global__ void elu_bias_kernel(float* __restrict__ x,
                                const float* __restrict__ b, int total, int C) {
  int i = blockIdx.x * blockDim.x + threadIdx.x;
  if (i >= total) return;
  float v = x[i] + b[i % C];
  x[i] = v > 0.f ? v : (expf(v) - 1.f);
}

__global__ void classifier_kernel(const float* __restrict__ h,
                                  const float* __restrict__ Wc,
                                  const float* __restrict__ bc,
                                  float* __restrict__ out, int N) {
  int n = blockIdx.x * blockDim.x + threadIdx.x;
  if (n >= N) return;
  float s0 = bc[0], s1 = bc[1];
  const float* row = h + (long)n * 64;
#pragma unroll 4
  for (int c = 0; c < 64; ++c) {
    float v = row[c];
    s0 += v * Wc[c * 2 + 0];
    s1 += v * Wc[c * 2 + 1];
  }
  out[n * 2 + 0] = s0;
  out[n * 2 + 1] = s1;
}

__global__ void fill_kernel(float* __restrict__ p, int n, float v) {
  int i = blockIdx.x * blockDim.x + threadIdx.x;
  if (i < n) p[i] = v;
}

static inline unsigned cdiv_u(long long a, long long b) {
  return (unsigned)((a + b - 1) / b);
}

extern "C" void kernel_launch(void* const* d_in, const int* in_sizes, int n_in,
                              void* d_out, int out_size, void* d_ws, size_t ws_size,
                              hipStream_t stream) {
  const float* x     = (const float*)d_in[0];
  const int*   ei    = (const int*)d_in[1];
  const float* W1    = (const float*)d_in[2];
  const float* a1s   = (const float*)d_in[3];
  const float* a1d   = (const float*)d_in[4];
  const float* b1    = (const float*)d_in[5];
  const float* W2    = (const float*)d_in[6];
  const float* a2s   = (const float*)d_in[7];
  const float* a2d   = (const float*)d_in[8];
  const float* b2    = (const float*)d_in[9];
  const float* Wc    = (const float*)d_in[10];
  const float* bc    = (const float*)d_in[11];
  float* out = (float*)d_out;

  const int N = in_sizes[0] / 8;     // 50000
  const int E = in_sizes[1] / 2;     // 800000
  const int ET = E + N;              // edges + self loops

  // workspace carving (floats): 660 * N total = ~132 MB
  float* ws   = (float*)d_ws;
  float* h1   = ws;                           // N*256
  float* out1 = h1   + (long)N * 256;         // N*256
  float* h2   = out1 + (long)N * 256;         // N*64
  float* out2 = h2   + (long)N * 64;          // N*64
  float* as1  = out2 + (long)N * 64;          // N*4
  float* ad1  = as1  + (long)N * 4;
  float* mx1  = ad1  + (long)N * 4;
  float* sm1  = mx1  + (long)N * 4;
  float* as2  = sm1  + (long)N * 4;           // N
  float* ad2  = as2  + N;
  float* mx2  = ad2  + N;
  float* sm2  = mx2  + N;

  const int TB = 256;
  const float NEG_INF = -__builtin_huge_valf();

  // ---- init accumulators ----
  fill_kernel<<<cdiv_u((long long)N * 256, TB), TB, 0, stream>>>(out1, N * 256, 0.f);
  fill_kernel<<<cdiv_u((long long)N * 4,   TB), TB, 0, stream>>>(mx1,  N * 4, NEG_INF);
  fill_kernel<<<cdiv_u((long long)N * 4,   TB), TB, 0, stream>>>(sm1,  N * 4, 0.f);
  fill_kernel<<<cdiv_u((long long)N * 64,  TB), TB, 0, stream>>>(out2, N * 64, 0.f);
  fill_kernel<<<cdiv_u(N, TB), TB, 0, stream>>>(mx2, N, NEG_INF);
  fill_kernel<<<cdiv_u(N, TB), TB, 0, stream>>>(sm2, N, 0.f);

  // ---- layer 1: h1 = x @ W1  (50000x8 @ 8x256) ----
  {
    long long waves = ((long long)N / 16) * (256 / 16);
    wmma_gemm_f32_k4<<<cdiv_u(waves * 32, TB), TB, 0, stream>>>(x, W1, h1, N, 8, 256);
  }
  alpha_kernel<<<cdiv_u((long long)N * 4, TB), TB, 0, stream>>>(h1, a1s, a1d, as1, ad1, N, 4);
  edge_max_kernel<<<cdiv_u(ET, TB), TB, 0, stream>>>(ei, E, N, as1, ad1, mx1, 4);
  edge_expsum_kernel<<<cdiv_u(ET, TB), TB, 0, stream>>>(ei, E, N, as1, ad1, mx1, sm1, 4);
  edge_aggregate_kernel<<<cdiv_u((long long)ET * 64, TB), TB, 0, stream>>>(
      ei, E, N, as1, ad1, mx1, sm1, h1, out1, 4);
  elu_bias_kernel<<<cdiv_u((long long)N * 256, TB), TB, 0, stream>>>(out1, b1, N * 256, 256);

  // ---- layer 2: h2 = out1 @ W2  (50000x256 @ 256x64) ----
  {
    long long waves = ((long long)N / 16) * (64 / 16);
    wmma_gemm_f32_k4<<<cdiv_u(waves * 32, TB), TB, 0, stream>>>(out1, W2, h2, N, 256, 64);
  }
  alpha_kernel<<<cdiv_u(N, TB), TB, 0, stream>>>(h2, a2s, a2d, as2, ad2, N, 1);
  edge_max_kernel<<<cdiv_u(ET, TB), TB, 0, stream>>>(ei, E, N, as2, ad2, mx2, 1);
  edge_expsum_kernel<<<cdiv_u(ET, TB), TB, 0, stream>>>(ei, E, N, as2, ad2, mx2, sm2, 1);
  edge_aggregate_kernel<<<cdiv_u((long long)ET * 64, TB), TB, 0, stream>>>(
      ei, E, N, as2, ad2, mx2, sm2, h2, out2, 1);
  elu_bias_kernel<<<cdiv_u((long long)N * 64, TB), TB, 0, stream>>>(out2, b2, N * 64, 64);

  // ---- classifier: out = out2 @ Wc + bc ----
  classifier_kernel<<<cdiv_u(N, TB), TB, 0, stream>>>(out2, Wc, bc, out, N);
}